// GCN_31061203485068
// MI455X (gfx1250) — compile-verified
//
#include <hip/hip_runtime.h>
#include <math.h>

// Problem constants (match reference)
#define NNODES 50000
#define NEDGES 800000
#define NFEAT  256
#define NHID   128
#define ZDIM   32
#define NPAIRS 10000

// Flip to 0 if the gfx1250 assembler rejects the async mnemonics.
#define USE_ASYNC_LDS 1

typedef __attribute__((ext_vector_type(2))) float v2f;
typedef __attribute__((ext_vector_type(4))) float v4f;
typedef __attribute__((ext_vector_type(8))) float v8f;

// ---------------------------------------------------------------------------
// CDNA5 async global->LDS copy (16B per lane), tracked by ASYNCcnt.
// LDS operand is a per-lane byte offset VGPR: generic shared pointers carry
// the LDS offset in their low 32 bits (ISA 10.2).
// ---------------------------------------------------------------------------
__device__ __forceinline__ void async_copy_b128(void* lds_dst, const void* gsrc) {
#if USE_ASYNC_LDS
  unsigned lds_off = (unsigned)(size_t)lds_dst;
  asm volatile("global_load_async_to_lds_b128 %0, %1, off"
               :: "v"(lds_off), "v"(gsrc)
               : "memory");
#else
  *(v4f*)lds_dst = *(const v4f*)gsrc;
#endif
}

__device__ __forceinline__ void wait_async_0() {
#if USE_ASYNC_LDS
  asm volatile("s_wait_asynccnt 0x0" ::: "memory");
#endif
}

// Partial wait: allow the newest 16 async loads (one ping-pong stage per wave)
// to remain in flight; older ones are guaranteed complete (in-order returns).
__device__ __forceinline__ void wait_async_16() {
#if USE_ASYNC_LDS
  asm volatile("s_wait_asynccnt 0x10" ::: "memory");
#endif
}

// ---------------------------------------------------------------------------
// row_ptr[i] = lower_bound(adj_row, i)   (adj_row is sorted)
// ---------------------------------------------------------------------------
__global__ void build_row_ptr(const int* __restrict__ row, int nnz, int n,
                              int* __restrict__ row_ptr) {
  int i = blockIdx.x * blockDim.x + threadIdx.x;
  if (i > n) return;
  int lo = 0, hi = nnz;
  while (lo < hi) {
    int mid = (lo + hi) >> 1;
    if (row[mid] < i) lo = mid + 1; else hi = mid;
  }
  row_ptr[i] = lo;
}

// ---------------------------------------------------------------------------
// C[M,NHID] = A[M,K] @ B[K,NHID] with V_WMMA_F32_16X16X4_F32.
// Block = 256 threads = 8 wave32s, computes a 128-row x 128-col tile:
//   - whole B panel [K x 128] async-staged into LDS once (<=128KB)
//   - wave w owns row stripe r0 = blk*128 + w*16, keeps 8 accumulators
//   - per K-step of 4: 1 global b64 A-load, 8 ds_load_2addr, 8 WMMAs
// Fragment layouts per ISA 7.12.2:
//   A 16x4:  lane l -> row (l&15), K = (l>>4)*2 + {0,1}
//   B 4x16:  lane l -> col (l&15), K = (l>>4)*2 + {0,1}
//   C 16x16: VGPR v, lane l -> row v + (l>>4)*8, col (l&15)
// ---------------------------------------------------------------------------
__global__ void gemm_wmma_f32(const float* __restrict__ A,
                              const float* __restrict__ B,
                              float* __restrict__ C, int M, int K) {
  extern __shared__ float Bp[];  // [K][NHID]

  const int tid  = threadIdx.x;
  const int wave = tid >> 5;
  const int lane = tid & 31;
  const int lm   = lane & 15;
  const int half = lane >> 4;
  const int kh   = half * 2;

  const int nvec = (K * NHID) / 4;
  for (int i = tid; i < nvec; i += 256)
    async_copy_b128(&Bp[i * 4], B + (size_t)i * 4);
  wait_async_0();
  __syncthreads();

  const int r0 = blockIdx.x * 128 + wave * 16;
  if (r0 < M) {  // wave-uniform; EXEC all-ones inside WMMA region
    const float* __restrict__ Arow = A + (size_t)(r0 + lm) * K;
    v8f acc[8];
#pragma unroll
    for (int t = 0; t < 8; ++t) acc[t] = (v8f){};

    for (int k = 0; k < K; k += 4) {
      if (k + 64 < K)
        __builtin_prefetch(Arow + k + 64, 0, 1);
      v2f a;
      a.x = Arow[k + kh];
      a.y = Arow[k + kh + 1];
      const float* __restrict__ Bk0 = &Bp[(k + kh) * NHID + lm];
#pragma unroll
      for (int t = 0; t < 8; ++t) {
        v2f b;
        b.x = Bk0[t * 16];
        b.y = Bk0[t * 16 + NHID];
        acc[t] = __builtin_amdgcn_wmma_f32_16x16x4_f32(
            false, a, false, b, (short)0, acc[t], false, false);
      }
    }
#pragma unroll
    for (int t = 0; t < 8; ++t)
#pragma unroll
      for (int v = 0; v < 8; ++v)
        C[(size_t)(r0 + v + half * 8) * NHID + t * 16 + lm] = acc[t][v];
  }
}

// ---------------------------------------------------------------------------
// x[n,:] = relu( sum_e val[e]*h[col[e],:] + bias ) (+ x[n,:] if use_res)
// One 128-thread block per node; edge segment from row_ptr (no atomics).
// ---------------------------------------------------------------------------
__global__ void spmm_epilogue(const int* __restrict__ row_ptr,
                              const int* __restrict__ col,
                              const float* __restrict__ val,
                              const float* __restrict__ h,
                              const float* __restrict__ bias,
                              float* __restrict__ x, int use_res) {
  const int n = blockIdx.x;
  const int t = threadIdx.x;  // 0..127
  const int e0 = row_ptr[n], e1 = row_ptr[n + 1];
  float acc = 0.f;
  for (int e = e0; e < e1; ++e)
    acc = fmaf(val[e], h[(size_t)col[e] * NHID + t], acc);
  float r = acc + bias[t];
  r = r > 0.f ? r : 0.f;
  if (use_res) r += x[(size_t)n * NHID + t];
  x[(size_t)n * NHID + t] = r;
}

// ---------------------------------------------------------------------------
// Decoder: per block, 16 pairs, double-buffered async staging of bil_w[k].
//   feat[p,k] = sum_e (A[p,:] @ bil_w[k])[e] * B[p,e]   (WMMA + shuffle red.)
//   pred[p]   = ELU(ELU(feat+bil_b) @ dw1 + db1) @ dw2 + db2
// Dynamic LDS: 2 x 64KB ping-pong for bil_w[k]; static: A/B tiles + feat.
// Per wave each stage = 16 async b128 issues -> s_wait_asynccnt 16 retires
// the previous stage while the new one stays in flight.
// ---------------------------------------------------------------------------
__global__ void decoder_kernel(const float* __restrict__ latent,
                               const int* __restrict__ idx,
                               const float* __restrict__ bil_w,
                               const float* __restrict__ bil_b,
                               const float* __restrict__ dw1,
                               const float* __restrict__ db1,
                               const float* __restrict__ dw2,
                               const float* __restrict__ db2,
                               float* __restrict__ pred) {
  extern __shared__ float Wping[];            // [2][NHID*NHID]
  __shared__ float Alds[16][NHID];
  __shared__ float Blds[16][NHID];
  __shared__ float feat[16][ZDIM];

  const int p0  = blockIdx.x * 16;
  const int tid = threadIdx.x;  // 0..255

  // Scalar ping-pong pointers (NOT an aggregate: an initialized pointer array
  // of LDS addrspacecasts becomes a static initializer ld.lld cannot link).
  float* const Wbuf0 = Wping;
  float* const Wbuf1 = Wping + NHID * NHID;

  // Kick off stage of bil_w[0] while we gather the pair tiles.
  {
    const float* src = bil_w;
    for (int i = tid; i < (NHID * NHID) / 4; i += 256)   // 16 issues/lane
      async_copy_b128(Wbuf0 + i * 4, src + (size_t)i * 4);
  }
  for (int i = tid; i < 16 * NHID; i += 256) {
    int p = i / NHID, d = i % NHID;
    Alds[p][d] = latent[(size_t)idx[p0 + p] * NHID + d];
    Blds[p][d] = latent[(size_t)idx[NPAIRS + p0 + p] * NHID + d];
  }
  for (int i = tid; i < 16 * ZDIM; i += 256) (&feat[0][0])[i] = 0.f;
  wait_async_0();
  __syncthreads();  // Wbuf0, tiles ready
  {
    const float* src = bil_w + (size_t)1 * NHID * NHID;
    for (int i = tid; i < (NHID * NHID) / 4; i += 256)
      async_copy_b128(Wbuf1 + i * 4, src + (size_t)i * 4);
  }

  const int wave = tid >> 5;
  const int lane = tid & 31;
  const int lm   = lane & 15;
  const int half = lane >> 4;
  const int kh   = half * 2;
  const int c0   = wave * 16;  // e-tile owned by this wave

  for (int k = 0; k < ZDIM; ++k) {
    const float* __restrict__ Wp = (k & 1) ? Wbuf1 : Wbuf0;
    v8f acc = {};
    for (int kk = 0; kk < NHID; kk += 4) {
      v2f a, b;
      a.x = Alds[lm][kk + kh];
      a.y = Alds[lm][kk + kh + 1];
      const float* __restrict__ Wk0 = &Wp[(kk + kh) * NHID + c0 + lm];
      b.x = Wk0[0];
      b.y = Wk0[NHID];
      acc = __builtin_amdgcn_wmma_f32_16x16x4_f32(false, a, false, b,
                                                  (short)0, acc, false, false);
    }
#pragma unroll
    for (int v = 0; v < 8; ++v) {
      int m = v + half * 8;  // pair row within tile
      float s = acc[v] * Blds[m][c0 + lm];
      s += __shfl_xor(s, 1, 32);
      s += __shfl_xor(s, 2, 32);
      s += __shfl_xor(s, 4, 32);
      s += __shfl_xor(s, 8, 32);
      if (lm == 0) atomicAdd(&feat[m][k], s);  // ds_add_f32 across 8 waves
    }
    __syncthreads();  // all waves done reading Wbuf[k&1]
    if (k + 2 < ZDIM) {
      const float* src = bil_w + (size_t)(k + 2) * NHID * NHID;
      float* dst = (k & 1) ? Wbuf1 : Wbuf0;
      for (int i = tid; i < (NHID * NHID) / 4; i += 256)
        async_copy_b128(dst + i * 4, src + (size_t)i * 4);
      wait_async_16();  // stage k+1 complete; stage k+2 still in flight
    } else {
      wait_async_0();   // drain tail
    }
    __syncthreads();    // publish Wbuf[(k+1)&1]
  }

  if (tid < 16) {
    const int p = tid;
    float f[ZDIM], hb[ZDIM];
#pragma unroll
    for (int k = 0; k < ZDIM; ++k) {
      float v = feat[p][k] + bil_b[k];
      f[k] = v > 0.f ? v : (expf(v) - 1.f);  // ELU
    }
    for (int j = 0; j < ZDIM; ++j) {
      float a = db1[j];
#pragma unroll
      for (int k = 0; k < ZDIM; ++k) a = fmaf(f[k], dw1[k * ZDIM + j], a);
      hb[j] = a > 0.f ? a : (expf(a) - 1.f);
    }
    float o = db2[0];
#pragma unroll
    for (int j = 0; j < ZDIM; ++j) o = fmaf(hb[j], dw2[j], o);
    pred[p0 + p] = o;
  }
}

// ---------------------------------------------------------------------------
extern "C" void kernel_launch(void* const* d_in, const int* in_sizes, int n_in,
                              void* d_out, int out_size, void* d_ws, size_t ws_size,
                              hipStream_t stream) {
  const float* features = (const float*)d_in[0];
  const int*   adj_row  = (const int*)d_in[1];
  const int*   adj_col  = (const int*)d_in[2];
  const float* adj_val  = (const float*)d_in[3];
  const int*   idx      = (const int*)d_in[4];
  const float* W0 = (const float*)d_in[5];
  const float* b0 = (const float*)d_in[6];
  const float* W1 = (const float*)d_in[7];
  const float* b1 = (const float*)d_in[8];
  const float* W2 = (const float*)d_in[9];
  const float* b2 = (const float*)d_in[10];
  const float* bil_w = (const float*)d_in[11];
  const float* bil_b = (const float*)d_in[12];
  const float* dw1 = (const float*)d_in[13];
  const float* db1 = (const float*)d_in[14];
  const float* dw2 = (const float*)d_in[15];
  const float* db2 = (const float*)d_in[16];

  float* pred   = (float*)d_out;          // [NPAIRS]
  float* latent = pred + NPAIRS;          // [NNODES*NHID], also running x

  float* h       = (float*)d_ws;                          // [NNODES*NHID]
  int*   row_ptr = (int*)(h + (size_t)NNODES * NHID);     // [NNODES+1]

  const int gemm_grid = (NNODES + 127) / 128;  // 391

  build_row_ptr<<<(NNODES + 1 + 255) / 256, 256, 0, stream>>>(
      adj_row, NEDGES, NNODES, row_ptr);

  // Layer 0: h = X @ W0 ; x = relu(spmm(h)+b0)
  gemm_wmma_f32<<<gemm_grid, 256, NFEAT * NHID * sizeof(float), stream>>>(
      features, W0, h, NNODES, NFEAT);
  spmm_epilogue<<<NNODES, NHID, 0, stream>>>(row_ptr, adj_col, adj_val, h, b0,
                                             latent, 0);
  // Layer 1: residual
  gemm_wmma_f32<<<gemm_grid, 256, NHID * NHID * sizeof(float), stream>>>(
      latent, W1, h, NNODES, NHID);
  spmm_epilogue<<<NNODES, NHID, 0, stream>>>(row_ptr, adj_col, adj_val, h, b1,
                                             latent, 1);
  // Layer 2: residual -> latent stays in d_out
  gemm_wmma_f32<<<gemm_grid, 256, NHID * NHID * sizeof(float), stream>>>(
      latent, W2, h, NNODES, NHID);
  spmm_epilogue<<<NNODES, NHID, 0, stream>>>(row_ptr, adj_col, adj_val, h, b2,
                                             latent, 1);

  // Decoder: 2 x 64KB dynamic LDS ping-pong for bil_w[k]
  decoder_kernel<<<NPAIRS / 16, 256, 2 * NHID * NHID * sizeof(float), stream>>>(
      latent, idx, bil_w, bil_b, dw1, db1, dw2, db2, pred);
}